// MiniBatchDiscrimination_15908558864539
// MI455X (gfx1250) — compile-verified
//
#include <hip/hip_runtime.h>
#include <math.h>

#define N_ROWS 1024
#define IN_F   2048
#define OUT_F  512

typedef float v2f __attribute__((ext_vector_type(2)));
typedef float v8f __attribute__((ext_vector_type(8)));

#define WMMA_F32(A, B, C)                                              \
  __builtin_amdgcn_wmma_f32_16x16x4_f32(false, (A), false, (B),        \
                                        (short)0, (C), false, false)

// ---------------------------------------------------------------------------
// Kernel 1: m[1024,512] = x[1024,2048] @ T[2048,512] via V_WMMA_F32_16X16X4_F32
// Each wave computes a 32x32 output tile = 2x2 WMMA tiles, stepping K by 4.
// Register blocking: per k-step, 2 A-operand pairs + 2 B-operand pairs feed
// 4 WMMAs -> 8 B/lane of operand traffic per WMMA (2x less than 1-tile/wave).
//
// A (16x4 f32) VGPR layout (ISA 7.12.2): lanes 0-15 hold M=lane with
//   VGPR0=K0, VGPR1=K1; lanes 16-31 hold M=lane-16 with VGPR0=K2, VGPR1=K3.
// B (4x16 f32) mirrored; C/D (16x16 f32): lanes 0-15 rows 0..7 in v0..7,
//   lanes 16-31 rows 8..15.
// ---------------------------------------------------------------------------
__global__ __launch_bounds__(128) void gemm_wmma_f32(
    const float* __restrict__ x, const float* __restrict__ T,
    float* __restrict__ m) {
  const int lane = threadIdx.x & 31;
  const int wave = threadIdx.x >> 5;
  const int tile = blockIdx.x * 4 + wave;          // 512 32x32 tiles total
  const int tiles_n = OUT_F / 32;                  // 16
  const int m0 = (tile / tiles_n) * 32;
  const int n0 = (tile % tiles_n) * 32;

  const int half = lane >> 4;                      // 0: lanes 0-15, 1: 16-31
  const int l = lane & 15;

  v8f acc00 = {}, acc01 = {}, acc10 = {}, acc11 = {};
  const float* arow0 = x + (size_t)(m0 + l) * IN_F + 2 * half;
  const float* arow1 = arow0 + (size_t)16 * IN_F;
  const float* bcol0 = T + (size_t)(2 * half) * OUT_F + (n0 + l);
  const float* bcol1 = bcol0 + 16;

#pragma unroll 2
  for (int k = 0; k < IN_F; k += 4) {
    v2f a0 = *(const v2f*)(arow0 + k);             // {x[m   ,k+2h], x[m   ,k+2h+1]}
    v2f a1 = *(const v2f*)(arow1 + k);             // {x[m+16,k+2h], x[m+16,k+2h+1]}
    v2f b0, b1;
    b0.x = bcol0[(size_t)k * OUT_F];               // T[k+2h  , n   ]
    b0.y = bcol0[(size_t)k * OUT_F + OUT_F];       // T[k+2h+1, n   ]
    b1.x = bcol1[(size_t)k * OUT_F];               // T[k+2h  , n+16]
    b1.y = bcol1[(size_t)k * OUT_F + OUT_F];       // T[k+2h+1, n+16]
    acc00 = WMMA_F32(a0, b0, acc00);
    acc01 = WMMA_F32(a0, b1, acc01);
    acc10 = WMMA_F32(a1, b0, acc10);
    acc11 = WMMA_F32(a1, b1, acc11);
  }

  // Store the four 16x16 tiles; lane half h covers rows M = 8h + r (r=0..7).
  float* o00 = m + (size_t)(m0 + 8 * half) * OUT_F + n0 + l;
  float* o10 = o00 + (size_t)16 * OUT_F;
#pragma unroll
  for (int r = 0; r < 8; ++r) {
    o00[(size_t)r * OUT_F]      = acc00[r];
    o00[(size_t)r * OUT_F + 16] = acc01[r];
    o10[(size_t)r * OUT_F]      = acc10[r];
    o10[(size_t)r * OUT_F + 16] = acc11[r];
  }
}

// ---------------------------------------------------------------------------
// Kernel 2: partial[i, chunk] = sum_{j in chunk} exp(-sum_f |m[i,f]-m[j,f]|)
// grid = (128, 4); block = 256 (8 waves). Wave w owns row i = bx*8 + w.
// j-chunk = 256 rows, staged 16 rows at a time in LDS (32 KB).
// Per lane: features f = k*32 + lane -> conflict-free LDS banks.
// global_prefetch of the NEXT j-tile overlaps its L2->WGP fill with compute.
// Wave32 shfl_xor tree reduce per j; partials (no float atomics) keep the
// result deterministic across graph replays.
// ---------------------------------------------------------------------------
#define JT 16
#define FPL (OUT_F / 32)   // 16 features per lane

__global__ __launch_bounds__(256) void pairwise_l1_exp(
    const float* __restrict__ m, float* __restrict__ part) {
  __shared__ float smj[JT * OUT_F];                // 32 KB
  const int lane = threadIdx.x & 31;
  const int wave = threadIdx.x >> 5;
  const int i = blockIdx.x * 8 + wave;             // 0..1023
  const int jbase = blockIdx.y * 256;              // 4 chunks of 256

  // Register copy of m[i] in lane-strided layout.
  float mi[FPL];
#pragma unroll
  for (int k = 0; k < FPL; ++k)
    mi[k] = m[(size_t)i * OUT_F + k * 32 + lane];

  float acc = 0.0f;
  for (int j0 = jbase; j0 < jbase + 256; j0 += JT) {
    __syncthreads();
    // Cooperative coalesced load of 16 j-rows (8192 floats / 256 threads).
#pragma unroll
    for (int t = 0; t < (JT * OUT_F) / 256; ++t) {
      int idx = t * 256 + threadIdx.x;
      smj[idx] = m[(size_t)j0 * OUT_F + idx];
    }
    __syncthreads();

    // Prefetch next tile (8192 floats = 256 x 128B lines, one per thread).
    if (j0 + JT < jbase + 256)
      __builtin_prefetch(m + (size_t)(j0 + JT) * OUT_F + threadIdx.x * 32, 0, 1);

#pragma unroll
    for (int jj = 0; jj < JT; ++jj) {
      const float* row = smj + jj * OUT_F;
      float d = 0.0f;
#pragma unroll
      for (int k = 0; k < FPL; ++k)
        d += fabsf(mi[k] - row[k * 32 + lane]);
      // wave32 all-reduce
#pragma unroll
      for (int off = 16; off > 0; off >>= 1)
        d += __shfl_xor(d, off, 32);
      acc += expf(-d);                             // same value in all lanes
    }
  }

  if (lane == 0) part[(size_t)i * 4 + blockIdx.y] = acc;
}

// ---------------------------------------------------------------------------
// Kernel 3: out[i, 0:2048] = x[i, :];  out[i, 2048] = sum of 4 partials.
// One block per row i.
// ---------------------------------------------------------------------------
__global__ __launch_bounds__(256) void finalize_concat(
    const float* __restrict__ x, const float* __restrict__ part,
    float* __restrict__ out) {
  const int i = blockIdx.x;
  const size_t orow = (size_t)i * (IN_F + 1);
  for (int f = threadIdx.x; f < IN_F; f += 256)
    out[orow + f] = x[(size_t)i * IN_F + f];
  if (threadIdx.x == 0) {
    const float* p = part + (size_t)i * 4;
    out[orow + IN_F] = p[0] + p[1] + p[2] + p[3];
  }
}

extern "C" void kernel_launch(void* const* d_in, const int* in_sizes, int n_in,
                              void* d_out, int out_size, void* d_ws, size_t ws_size,
                              hipStream_t stream) {
  const float* x = (const float*)d_in[0];   // [1024, 2048]
  const float* T = (const float*)d_in[1];   // [2048, 512]
  float* out = (float*)d_out;               // [1024, 2049]

  float* m    = (float*)d_ws;                               // 1024*512 floats (2 MB)
  float* part = m + (size_t)N_ROWS * OUT_F;                 // 1024*4 floats

  // 1) GEMM: 512 32x32 tiles, 4 waves/block -> 128 blocks.
  gemm_wmma_f32<<<128, 128, 0, stream>>>(x, T, m);

  // 2) Pairwise L1 + exp partial sums: (128 i-blocks) x (4 j-chunks).
  dim3 g2(128, 4);
  pairwise_l1_exp<<<g2, 256, 0, stream>>>(m, part);

  // 3) Concat x with reduced feature column.
  finalize_concat<<<N_ROWS, 256, 0, stream>>>(x, part, out);
}